// Attention_20340965114245
// MI455X (gfx1250) — compile-verified
//
#include <hip/hip_runtime.h>
#include <hip/hip_bf16.h>
#include <cstddef>

// ---------------------------------------------------------------------------
// GPT-2 attention forward for gfx1250 (MI455X), fp32 via V_WMMA_F32_16X16X4_F32
// B=2, S=2048, D=1024, H=16, hd=64
//
// LDS tiles are stored pair-interleaved along K:
//   lX2[j][i] = { X[2j][i], X[2j+1][i] }  (float2)
// so every WMMA A/B fragment is one aligned ds_load_b64:
//   frag(lane) = lX2[ kk/2 + (lane<16 ? 0 : 1) ][ i(lane) ]
// Row pad P ≡ 16 (mod 32)  =>  2P ≡ 32 (mod 64): the two half-waves hit
// disjoint 32-bank sets -> conflict-free b64 reads.
// ---------------------------------------------------------------------------

typedef __attribute__((ext_vector_type(2))) float v2f;
typedef __attribute__((ext_vector_type(8))) float v8f;

__device__ __forceinline__ v8f wmma4(v2f a, v2f b, v8f c) {
  return __builtin_amdgcn_wmma_f32_16x16x4_f32(
      /*neg_a=*/false, a, /*neg_b=*/false, b,
      /*c_mod=*/(short)0, c, /*reuse_a=*/false, /*reuse_b=*/false);
}

__device__ __forceinline__ v8f vzero8() {
  v8f z = {0.f, 0.f, 0.f, 0.f, 0.f, 0.f, 0.f, 0.f};
  return z;
}

// ---------------------------------------------------------------------------
// Generic row-major GEMM + bias:  C[M,N] = A[M,K] @ B[K,N] + bias[N]
// Block tile 64(M) x 128(N), BK=16. 256 thr = 8 waves in 4x2 grid,
// each wave computes 16x64 via 4 f32 WMMA accumulators.
// Grid: x = N/128, y = M/64.
// ---------------------------------------------------------------------------
__global__ __launch_bounds__(256)
void gemm_bias_kernel(const float* __restrict__ A, const float* __restrict__ B,
                      const float* __restrict__ bias, float* __restrict__ C,
                      int K, int lda, int ldb, int ldc) {
  constexpr int PA = 80;    // 64 + 16
  constexpr int PB = 144;   // 128 + 16
  __shared__ v2f lA2[8 * PA];   // [j][m] = {A[m][2j], A[m][2j+1]}
  __shared__ v2f lB2[8 * PB];   // [j][n] = {B[2j][n], B[2j+1][n]}

  const int tid  = threadIdx.x;
  const int lane = tid & 31;
  const int wid  = tid >> 5;
  const int wm   = wid & 3;       // 0..3  -> 16-row slice
  const int wn   = wid >> 2;      // 0..1  -> 64-col slice
  const int m0   = blockIdx.y * 64;
  const int n0   = blockIdx.x * 128;

  v8f acc[4] = {vzero8(), vzero8(), vzero8(), vzero8()};

  for (int k0 = 0; k0 < K; k0 += 16) {
    // stage A tile 64x16 (thread: one float4 along K -> two paired ds_store_b64)
    {
      const int r = tid >> 2;            // 0..63 (m)
      const int c = (tid & 3) * 4;       // 0,4,8,12 (k offset)
      const float4 v = *(const float4*)(A + (size_t)(m0 + r) * lda + k0 + c);
      v2f p0; p0.x = v.x; p0.y = v.y;
      v2f p1; p1.x = v.z; p1.y = v.w;
      lA2[(c / 2 + 0) * PA + r] = p0;
      lA2[(c / 2 + 1) * PA + r] = p1;
    }
    // stage B tile 16x128: thread loads rows 2j,2j+1 and writes K-interleaved
    {
      const int j = tid >> 5;            // 0..7 (k pair)
      const int c = (tid & 31) * 4;      // 0..124 (n offset)
      const float4 b0 = *(const float4*)(B + (size_t)(k0 + 2 * j) * ldb + n0 + c);
      const float4 b1 = *(const float4*)(B + (size_t)(k0 + 2 * j + 1) * ldb + n0 + c);
      float4 w0; w0.x = b0.x; w0.y = b1.x; w0.z = b0.y; w0.w = b1.y;
      float4 w1; w1.x = b0.z; w1.y = b1.z; w1.z = b0.w; w1.w = b1.w;
      *(float4*)(&lB2[(size_t)j * PB + c])     = w0;
      *(float4*)(&lB2[(size_t)j * PB + c + 2]) = w1;
    }
    if (k0 + 16 < K) {                   // speculative prefetch of next tiles
      __builtin_prefetch(A + (size_t)(m0 + (tid >> 2)) * lda + k0 + 16, 0, 0);
      __builtin_prefetch(B + (size_t)(k0 + 16 + (tid >> 5)) * ldb + n0 + (tid & 31) * 4, 0, 0);
    }
    __syncthreads();

    const int joff = (lane < 16) ? 0 : 1;
    const int ml   = wm * 16 + (lane & 15);
#pragma unroll
    for (int kk2 = 0; kk2 < 8; kk2 += 2) {     // K pairs: 2 per WMMA slab of 4
      const v2f a = lA2[(kk2 + joff) * PA + ml];
#pragma unroll
      for (int nt = 0; nt < 4; nt++) {
        const int nl = wn * 64 + nt * 16 + (lane & 15);
        const v2f b = lB2[(kk2 + joff) * PB + nl];
        acc[nt] = wmma4(a, b, acc[nt]);
      }
    }
    __syncthreads();
  }

  // epilogue: C/D layout — vgpr r holds M = r (+8 for lanes 16-31), N = lane%16
  const int rbase = m0 + wm * 16 + ((lane < 16) ? 0 : 8);
  const int cl    = lane & 15;
#pragma unroll
  for (int nt = 0; nt < 4; nt++) {
    const int col = n0 + wn * 64 + nt * 16 + cl;
    const float bv = bias[col];
#pragma unroll
    for (int r = 0; r < 8; r++) {
      C[(size_t)(rbase + r) * ldc + col] = acc[nt][r] + bv;
    }
  }
}

// ---------------------------------------------------------------------------
// present[2,B,H,S,hd]: present[0]=K, present[1]=V, re-gathered from qkv.
// ---------------------------------------------------------------------------
__global__ __launch_bounds__(256)
void present_kernel(const float* __restrict__ qkv, float* __restrict__ present) {
  const int idx = blockIdx.x * 256 + threadIdx.x;  // float4 index, 2^21 total
  const int d4 = idx & 15;
  const int s  = (idx >> 4) & 2047;
  const int h  = (idx >> 15) & 15;
  const int b  = (idx >> 19) & 1;
  const int kv = idx >> 20;
  const float4 v = *(const float4*)(qkv + (size_t)(b * 2048 + s) * 3072 +
                                    (size_t)(1 + kv) * 1024 + h * 64 + d4 * 4);
  ((float4*)present)[idx] = v;
}

// ---------------------------------------------------------------------------
// Causal scores: out[q,k] = (Q·K)/8 if k<=q else -10000.  Per (b,h).
// A = Q (lda=3072); B = K^T built by K-pair-interleaved transpose into LDS.
// Grid: x = S/128, y = S/64, z = B*H.
// ---------------------------------------------------------------------------
__global__ __launch_bounds__(256)
void scores_kernel(const float* __restrict__ qkv, float* __restrict__ attn) {
  const int S = 2048, TD = 3072, D = 1024, hd = 64;
  constexpr int PA = 80;    // 64 + 16
  constexpr int PB = 144;   // 128 + 16
  const int bh = blockIdx.z;
  const int b = bh >> 4, h = bh & 15;
  const int m0 = blockIdx.y * 64;
  const int n0 = blockIdx.x * 128;
  float* out = attn + (size_t)bh * S * S;
  const int tid = threadIdx.x;

  if (n0 > m0 + 63) {  // tile entirely above the diagonal: all masked
    const float4 mv = make_float4(-10000.f, -10000.f, -10000.f, -10000.f);
#pragma unroll
    for (int i = 0; i < 8; i++) {
      const int idx = tid + i * 256;     // float4 index within 64x128 tile
      const int r = idx >> 5;            // 32 float4 per row
      const int c = (idx & 31) * 4;
      *(float4*)(out + (size_t)(m0 + r) * S + n0 + c) = mv;
    }
    return;
  }

  const float* Qb = qkv + (size_t)b * S * TD + h * hd;       // Q[q][d]
  const float* Kb = qkv + (size_t)b * S * TD + D + h * hd;   // K[k][d]

  __shared__ v2f lA2[8 * PA];
  __shared__ v2f lB2[8 * PB];
  const int lane = tid & 31;
  const int wid  = tid >> 5;
  const int wm   = wid & 3;
  const int wn   = wid >> 2;

  v8f acc[4] = {vzero8(), vzero8(), vzero8(), vzero8()};

  for (int k0 = 0; k0 < 64; k0 += 16) {
    {  // A: Q tile 64x16
      const int r = tid >> 2;
      const int c = (tid & 3) * 4;
      const float4 v = *(const float4*)(Qb + (size_t)(m0 + r) * TD + k0 + c);
      v2f p0; p0.x = v.x; p0.y = v.y;
      v2f p1; p1.x = v.z; p1.y = v.w;
      lA2[(c / 2 + 0) * PA + r] = p0;
      lA2[(c / 2 + 1) * PA + r] = p1;
    }
    {  // B: K^T tile 16x128 — transpose; d-contiguous loads become K pairs
      const int n  = tid >> 1;           // 0..127 (key index within tile)
      const int kg = (tid & 1) * 8;      // 0 or 8 (d-offset)
      const float4 v0 = *(const float4*)(Kb + (size_t)(n0 + n) * TD + k0 + kg);
      const float4 v1 = *(const float4*)(Kb + (size_t)(n0 + n) * TD + k0 + kg + 4);
      v2f p;
      p.x = v0.x; p.y = v0.y; lB2[(kg / 2 + 0) * PB + n] = p;
      p.x = v0.z; p.y = v0.w; lB2[(kg / 2 + 1) * PB + n] = p;
      p.x = v1.x; p.y = v1.y; lB2[(kg / 2 + 2) * PB + n] = p;
      p.x = v1.z; p.y = v1.w; lB2[(kg / 2 + 3) * PB + n] = p;
    }
    __syncthreads();

    const int joff = (lane < 16) ? 0 : 1;
    const int ml   = wm * 16 + (lane & 15);
#pragma unroll
    for (int kk2 = 0; kk2 < 8; kk2 += 2) {
      const v2f a = lA2[(kk2 + joff) * PA + ml];
#pragma unroll
      for (int nt = 0; nt < 4; nt++) {
        const int nl = wn * 64 + nt * 16 + (lane & 15);
        const v2f bb = lB2[(kk2 + joff) * PB + nl];
        acc[nt] = wmma4(a, bb, acc[nt]);
      }
    }
    __syncthreads();
  }

  const int rbase = m0 + wm * 16 + ((lane < 16) ? 0 : 8);
  const int cl    = lane & 15;
#pragma unroll
  for (int nt = 0; nt < 4; nt++) {
    const int col = n0 + wn * 64 + nt * 16 + cl;
#pragma unroll
    for (int r = 0; r < 8; r++) {
      const int row = rbase + r;
      const float v = (col <= row) ? acc[nt][r] * 0.125f : -10000.0f;
      out[(size_t)row * S + col] = v;
    }
  }
}

// ---------------------------------------------------------------------------
// In-place row softmax over S=2048. One block per row; 8 elems/thread.
// ---------------------------------------------------------------------------
__global__ __launch_bounds__(256)
void softmax_kernel(float* __restrict__ attn) {
  float* p = attn + (size_t)blockIdx.x * 2048;
  const int tid = threadIdx.x;
  float4* p4 = (float4*)p;
  float4 v0 = p4[tid];
  float4 v1 = p4[tid + 256];

  __shared__ float red[256];
  float m = fmaxf(fmaxf(fmaxf(v0.x, v0.y), fmaxf(v0.z, v0.w)),
                  fmaxf(fmaxf(v1.x, v1.y), fmaxf(v1.z, v1.w)));
  red[tid] = m;
  __syncthreads();
  for (int s = 128; s > 0; s >>= 1) {
    if (tid < s) red[tid] = fmaxf(red[tid], red[tid + s]);
    __syncthreads();
  }
  m = red[0];
  __syncthreads();

  v0.x = __expf(v0.x - m); v0.y = __expf(v0.y - m);
  v0.z = __expf(v0.z - m); v0.w = __expf(v0.w - m);
  v1.x = __expf(v1.x - m); v1.y = __expf(v1.y - m);
  v1.z = __expf(v1.z - m); v1.w = __expf(v1.w - m);
  float sum = v0.x + v0.y + v0.z + v0.w + v1.x + v1.y + v1.z + v1.w;
  red[tid] = sum;
  __syncthreads();
  for (int s = 128; s > 0; s >>= 1) {
    if (tid < s) red[tid] += red[tid + s];
    __syncthreads();
  }
  const float inv = 1.0f / red[0];

  v0.x *= inv; v0.y *= inv; v0.z *= inv; v0.w *= inv;
  v1.x *= inv; v1.y *= inv; v1.z *= inv; v1.w *= inv;
  p4[tid] = v0;
  p4[tid + 256] = v1;
}

// ---------------------------------------------------------------------------
// a = attn @ V per (b,h). Block tile 128(M) x 64(N), 8 waves stacked on M.
// Causal truncation: keys >= m0+128 contribute exactly 0 -> skip.
// Grid: x = S/128, y = B*H. Output merged-head layout [B,S,D].
// ---------------------------------------------------------------------------
__global__ __launch_bounds__(256)
void av_kernel(const float* __restrict__ attn, const float* __restrict__ qkv,
               float* __restrict__ amerged) {
  const int S = 2048, TD = 3072, D = 1024, hd = 64;
  constexpr int PA = 144;   // 128 + 16
  constexpr int PB = 80;    //  64 + 16
  const int bh = blockIdx.y;
  const int b = bh >> 4, h = bh & 15;
  const int m0 = blockIdx.x * 128;
  const float* A = attn + (size_t)bh * S * S;                     // [S,S]
  const float* V = qkv + (size_t)b * S * TD + 2 * D + h * hd;     // V[k][d]
  float* C = amerged + (size_t)b * S * D + h * hd;                // [S,hd] slice

  __shared__ v2f lA2[8 * PA];
  __shared__ v2f lB2[8 * PB];
  const int tid  = threadIdx.x;
  const int lane = tid & 31;
  const int wid  = tid >> 5;   // 0..7 -> 16-row slice of 128

  v8f acc[4] = {vzero8(), vzero8(), vzero8(), vzero8()};

  const int kmax = m0 + 128;   // causal: attn[q,k]==0 for k>q
  for (int k0 = 0; k0 < kmax; k0 += 16) {
    {  // A tile 128x16: thread loads 8 K-contiguous -> 4 paired ds_store_b64
      const int r = tid >> 1;           // 0..127 (m)
      const int c = (tid & 1) * 8;      // 0 or 8 (k offset)
      const float4 a0 = *(const float4*)(A + (size_t)(m0 + r) * S + k0 + c);
      const float4 a1 = *(const float4*)(A + (size_t)(m0 + r) * S + k0 + c + 4);
      v2f p;
      p.x = a0.x; p.y = a0.y; lA2[(c / 2 + 0) * PA + r] = p;
      p.x = a0.z; p.y = a0.w; lA2[(c / 2 + 1) * PA + r] = p;
      p.x = a1.x; p.y = a1.y; lA2[(c / 2 + 2) * PA + r] = p;
      p.x = a1.z; p.y = a1.w; lA2[(c / 2 + 3) * PA + r] = p;
    }
    {  // B tile 16x64: thread loads rows 2j,2j+1 (2 cols) -> one b128 store
      const int j = tid >> 5;           // 0..7 (k pair)
      const int c = (tid & 31) * 2;     // 0..62 (n offset)
      const float2 r0 = *(const float2*)(V + (size_t)(k0 + 2 * j) * TD + c);
      const float2 r1 = *(const float2*)(V + (size_t)(k0 + 2 * j + 1) * TD + c);
      float4 w; w.x = r0.x; w.y = r1.x; w.z = r0.y; w.w = r1.y;
      *(float4*)(&lB2[(size_t)j * PB + c]) = w;
    }
    __syncthreads();

    const int joff = (lane < 16) ? 0 : 1;
    const int ml   = wid * 16 + (lane & 15);
#pragma unroll
    for (int kk2 = 0; kk2 < 8; kk2 += 2) {
      const v2f a = lA2[(kk2 + joff) * PA + ml];
#pragma unroll
      for (int nt = 0; nt < 4; nt++) {
        const int nl = nt * 16 + (lane & 15);
        const v2f bb = lB2[(kk2 + joff) * PB + nl];
        acc[nt] = wmma4(a, bb, acc[nt]);
      }
    }
    __syncthreads();
  }

  const int rbase = m0 + wid * 16 + ((lane < 16) ? 0 : 8);
  const int cl    = lane & 15;
#pragma unroll
  for (int nt = 0; nt < 4; nt++) {
    const int col = nt * 16 + cl;
#pragma unroll
    for (int r = 0; r < 8; r++) {
      C[(size_t)(rbase + r) * D + col] = acc[nt][r];
    }
  }
}

// ---------------------------------------------------------------------------
// Host launcher
// ---------------------------------------------------------------------------
extern "C" void kernel_launch(void* const* d_in, const int* in_sizes, int n_in,
                              void* d_out, int out_size, void* d_ws, size_t ws_size,
                              hipStream_t stream) {
  (void)in_sizes; (void)n_in; (void)out_size; (void)ws_size;
  const float* x      = (const float*)d_in[0];  // [2,2048,1024]
  const float* w_attn = (const float*)d_in[1];  // [1024,3072]
  const float* b_attn = (const float*)d_in[2];  // [3072]
  const float* w_proj = (const float*)d_in[3];  // [1024,1024]
  const float* b_proj = (const float*)d_in[4];  // [1024]

  float* out_a       = (float*)d_out;                         // 2*2048*1024
  float* out_present = out_a + (size_t)4194304;               // 2*2*16*2048*64
  float* out_attn    = out_present + (size_t)8388608;         // 2*16*2048*2048

  float* qkv     = (float*)d_ws;                              // [4096,3072] f32
  float* amerged = qkv + (size_t)12582912;                    // [4096,1024] f32

  // 1) qkv = x @ w_attn + b_attn     (M=4096, N=3072, K=1024)
  gemm_bias_kernel<<<dim3(24, 64), 256, 0, stream>>>(
      x, w_attn, b_attn, qkv, 1024, 1024, 3072, 3072);

  // 2) present = stack(k, v)
  present_kernel<<<8192, 256, 0, stream>>>(qkv, out_present);

  // 3) raw causal scores into the attn output region
  scores_kernel<<<dim3(16, 32, 32), 256, 0, stream>>>(qkv, out_attn);

  // 4) softmax in place (B*H*S rows)
  softmax_kernel<<<65536, 256, 0, stream>>>(out_attn);

  // 5) a_heads = attn @ v  (merged-head layout)
  av_kernel<<<dim3(16, 32), 256, 0, stream>>>(out_attn, qkv, amerged);

  // 6) out = a_merged @ w_proj + b_proj   (M=4096, N=1024, K=1024)
  gemm_bias_kernel<<<dim3(8, 64), 256, 0, stream>>>(
      amerged, w_proj, b_proj, out_a, 1024, 1024, 1024, 1024);
}